// GSNN_26018911879782
// MI455X (gfx1250) — compile-verified
//
#include <hip/hip_runtime.h>

// Sizes (fixed by the reference problem)
#define NN   10000
#define II   1000
#define FF   8600
#define EE   80000
#define CC   16
#define BB   64
#define NLAYERS 6
#define EPSL 1e-5f
#define NB3 (EE/16)   // edge-kernel block count (16 edges per block)

typedef __attribute__((ext_vector_type(2))) float v2f;
typedef __attribute__((ext_vector_type(8))) float v8f;

__device__ __forceinline__ float eluf(float v) {
  return v > 0.0f ? v : (expf(v) - 1.0f);
}

// ---------------- one-time prep kernels ----------------

__global__ void k_zero2(int* a, int* b, int n) {
  int i = blockIdx.x * blockDim.x + threadIdx.x;
  if (i < n) { a[i] = 0; b[i] = 0; }
}

// w2t[f][d][c] = w2[f][c][d]  (B-matrix KxN columns become contiguous, f32)
__global__ void k_w2t(const float* __restrict__ w2, float* __restrict__ w2t) {
  int i = blockIdx.x * blockDim.x + threadIdx.x;
  if (i >= FF * 256) return;
  int f = i >> 8, r = i & 255, d = r >> 4, c = r & 15;
  w2t[(size_t)f * 256 + d * 16 + c] = w2[(size_t)f * 256 + c * 16 + d];
}

// x0[e][b] = xe[e][b] = x[b][src[e]]
__global__ void k_x0(const float* __restrict__ x, const int* __restrict__ src,
                     float* __restrict__ x0, float* __restrict__ xe) {
  int i = blockIdx.x * blockDim.x + threadIdx.x;
  if (i >= EE * BB) return;
  int e = i >> 6, b = i & 63;
  float v = x[(size_t)b * NN + src[e]];
  x0[i] = v; xe[i] = v;
}

__global__ void k_count(const int* __restrict__ dst, int* __restrict__ cnt) {
  int e = blockIdx.x * blockDim.x + threadIdx.x;
  if (e >= EE) return;
  int d = dst[e];
  if (d < II + FF) atomicAdd(&cnt[d - II], 1);
}

// single-block exclusive scan over FF counts -> offsets, off[FF] = total
__global__ void k_scan(const int* __restrict__ cnt, int* __restrict__ off) {
  __shared__ int part[257];
  int t = threadIdx.x;
  const int chunk = (FF + 255) / 256;
  int lo = t * chunk, hi = lo + chunk; if (hi > FF) hi = FF;
  int s = 0;
  for (int i = lo; i < hi; ++i) s += cnt[i];
  part[t] = s;
  __syncthreads();
  if (t == 0) {
    int run = 0;
    for (int i = 0; i < 256; ++i) { int v = part[i]; part[i] = run; run += v; }
    off[FF] = run;
  }
  __syncthreads();
  int run = part[t];
  for (int i = lo; i < hi; ++i) { off[i] = run; run += cnt[i]; }
}

__global__ void k_fill(const int* __restrict__ dst, const int* __restrict__ off,
                       int* __restrict__ cursor, int* __restrict__ edges) {
  int e = blockIdx.x * blockDim.x + threadIdx.x;
  if (e >= EE) return;
  int d = dst[e];
  if (d < II + FF) {
    int f = d - II;
    int p = atomicAdd(&cursor[f], 1);
    edges[off[f] + p] = e;
  }
}

// per-bin insertion sort -> deterministic summation order
__global__ void k_sort(const int* __restrict__ off, int* __restrict__ edges) {
  int f = blockIdx.x * blockDim.x + threadIdx.x;
  if (f >= FF) return;
  int beg = off[f], end = off[f + 1];
  for (int i = beg + 1; i < end; ++i) {
    int v = edges[i], j = i - 1;
    while (j >= beg && edges[j] > v) { edges[j + 1] = edges[j]; --j; }
    edges[j + 1] = v;
  }
}

// ---------------- per-layer kernels ----------------

// h[f][b][c] = elu( b1[I+f][c] + sum_{e in CSR(f)} xe[e][b]*w1[e][c] )
__global__ void k_h(const float* __restrict__ xe, const float* __restrict__ w1,
                    const float* __restrict__ b1, const int* __restrict__ off,
                    const int* __restrict__ edges, float* __restrict__ h) {
  int f = blockIdx.x;
  int t = threadIdx.x;          // 256: b = t>>2 (0..63), 4 channels per thread
  int b = t >> 2;
  int cq = (t & 3) << 2;
  float a0 = 0.f, a1 = 0.f, a2 = 0.f, a3 = 0.f;
  int beg = off[f], end = off[f + 1];
  for (int i = beg; i < end; ++i) {
    int e = edges[i];
    float xv = xe[(size_t)e * BB + b];
    float4 w = *(const float4*)(w1 + (size_t)e * CC + cq);
    a0 += xv * w.x; a1 += xv * w.y; a2 += xv * w.z; a3 += xv * w.w;
  }
  float4 bb = *(const float4*)(b1 + (size_t)(II + f) * CC + cq);
  float4 r;
  r.x = eluf(a0 + bb.x); r.y = eluf(a1 + bb.y);
  r.z = eluf(a2 + bb.z); r.w = eluf(a3 + bb.w);
  *(float4*)(h + ((size_t)f * BB + b) * CC + cq) = r;
}

// One wave per function node: A = h[f] (64x16 batch x C, f32), B = w2t[f] (KxN
// f32). K=16 -> chain of 4 x v_wmma_f32_16x16x4_f32 per 16-row M-tile, 4 tiles.
// Exact f32 precision, no conversions. In-place h -> h2 = elu(A@B + b2).
//
// 32-bit A 16x4 layout: lane L holds (M = L&15, K = 2*(L>=16) + {0,1}) -> per
// K-block kb a float2 at row offset 4*kb + 2*half.
// 32-bit B 4x16 layout: VGPR j holds (K = j + 2*(L>=16), N = L&15) -> per
// K-block a float2 from the transposed w2t row d=L&15.
// f32 C/D 16x16 layout: VGPR v -> (M = v + 8*(L>=16), N = L&15).
__global__ void k_wmma(float* h, const float* __restrict__ w2t,
                       const float* __restrict__ b2) {
  int lane = threadIdx.x & 31;
  int f = blockIdx.x * 8 + (threadIdx.x >> 5);   // 8 waves / block, FF % 8 == 0
  int half = lane >> 4;
  int l16  = lane & 15;

  v2f bm[4];
  const float* wrow = w2t + (size_t)f * 256 + l16 * 16 + 2 * half;
#pragma unroll
  for (int kb = 0; kb < 4; ++kb)
    bm[kb] = *(const v2f*)(wrow + 4 * kb);

  v2f am[4][4];
#pragma unroll
  for (int mt = 0; mt < 4; ++mt) {
    const float* hrow = h + ((size_t)f * BB + mt * 16 + l16) * CC + 2 * half;
#pragma unroll
    for (int kb = 0; kb < 4; ++kb)
      am[mt][kb] = *(const v2f*)(hrow + 4 * kb);
  }

  v8f acc[4];
#pragma unroll
  for (int mt = 0; mt < 4; ++mt) {
    v8f c = {};
#pragma unroll
    for (int kb = 0; kb < 4; ++kb)
      c = __builtin_amdgcn_wmma_f32_16x16x4_f32(false, am[mt][kb], false, bm[kb],
                                                (short)0, c, false, false);
    acc[mt] = c;
  }

  float bb = b2[(size_t)(II + f) * CC + l16];
#pragma unroll
  for (int mt = 0; mt < 4; ++mt) {
#pragma unroll
    for (int v = 0; v < 8; ++v) {
      int m = mt * 16 + v + half * 8;
      h[((size_t)f * BB + m) * CC + l16] = eluf(acc[mt][v] + bb);
    }
  }
}

// xe_pre[e][b] = xe + b3[e] + (src>=I ? dot(h2[src-I][b], w3[e]) : 0)
// (input-src edges have w3 == 0 structurally); accumulate block partial sums.
__global__ void k_edge(const float* __restrict__ h2, const float* __restrict__ w3,
                       const float* __restrict__ b3, const int* __restrict__ src,
                       float* __restrict__ xe, float* __restrict__ partial) {
  __shared__ float ss[256], sq[256];
  int t = threadIdx.x;
  int b = t & 63;
  int g = t >> 6;       // 0..3
  float ls = 0.f, lq = 0.f;
#pragma unroll
  for (int i = 0; i < 4; ++i) {
    int e = blockIdx.x * 16 + g * 4 + i;
    int sf = src[e];
    float pre = xe[(size_t)e * BB + b] + b3[e];
    if (sf >= II) {
      const float4* hp = (const float4*)(h2 + ((size_t)(sf - II) * BB + b) * CC);
      const float4* wp = (const float4*)(w3 + (size_t)e * CC);
      float d = 0.f;
#pragma unroll
      for (int j = 0; j < 4; ++j) {
        float4 a = hp[j], w = wp[j];
        d += a.x * w.x + a.y * w.y + a.z * w.z + a.w * w.w;
      }
      pre += d;
    }
    xe[(size_t)e * BB + b] = pre;
    ls += pre; lq += pre * pre;
  }
  ss[t] = ls; sq[t] = lq;
  __syncthreads();
  if (t < 64) {
    float s = ss[t] + ss[t + 64] + ss[t + 128] + ss[t + 192];
    float q = sq[t] + sq[t + 64] + sq[t + 128] + sq[t + 192];
    partial[(size_t)b * NB3 + blockIdx.x] = s;
    partial[(size_t)64 * NB3 + (size_t)b * NB3 + blockIdx.x] = q;
  }
}

// per-batch mean / rstd over E edges (fixed-order tree -> deterministic)
__global__ void k_stats(const float* __restrict__ partial, float* __restrict__ stats) {
  __shared__ float ss[256], sq[256];
  int b = blockIdx.x, t = threadIdx.x;
  float s = 0.f, q = 0.f;
  for (int i = t; i < NB3; i += 256) {
    s += partial[(size_t)b * NB3 + i];
    q += partial[(size_t)64 * NB3 + (size_t)b * NB3 + i];
  }
  ss[t] = s; sq[t] = q;
  __syncthreads();
  for (int st = 128; st > 0; st >>= 1) {
    if (t < st) { ss[t] += ss[t + st]; sq[t] += sq[t + st]; }
    __syncthreads();
  }
  if (t == 0) {
    float mu  = ss[0] * (1.0f / EE);
    float var = sq[0] * (1.0f / EE) - mu * mu;
    stats[b * 2]     = mu;
    stats[b * 2 + 1] = rsqrtf(var + EPSL);
  }
}

// xe = (1-m)*((pre-mu)*rstd) + m*x0
__global__ void k_norm(const float* __restrict__ stats, const float* __restrict__ mask,
                       const float* __restrict__ x0, float* __restrict__ xe) {
  int i = blockIdx.x * blockDim.x + threadIdx.x;
  if (i >= EE * BB) return;
  int e = i >> 6, b = i & 63;
  float m  = mask[e];
  float xn = (xe[i] - stats[b * 2]) * stats[b * 2 + 1];
  xe[i] = (1.0f - m) * xn + m * x0[i];
}

// out[b][n] = (scale[last_edge[n]]*xe[last_edge[n]][b] + bias[...]) * has_edge[n]
__global__ void k_out(const float* __restrict__ xe, const float* __restrict__ scale,
                      const float* __restrict__ bias, const int* __restrict__ last_edge,
                      const float* __restrict__ has_edge, float* __restrict__ out) {
  int i = blockIdx.x * blockDim.x + threadIdx.x;
  if (i >= BB * NN) return;
  int b = i / NN, n = i - b * NN;
  int e = last_edge[n];
  out[i] = (scale[e] * xe[(size_t)e * BB + b] + bias[e]) * has_edge[n];
}

// ---------------- launcher ----------------

extern "C" void kernel_launch(void* const* d_in, const int* in_sizes, int n_in,
                              void* d_out, int out_size, void* d_ws, size_t ws_size,
                              hipStream_t stream) {
  (void)in_sizes; (void)n_in; (void)out_size; (void)ws_size;
  const float* x    = (const float*)d_in[0];
  const float* w1   = (const float*)d_in[1];
  const float* b1   = (const float*)d_in[2];
  const float* w2   = (const float*)d_in[3];
  const float* b2   = (const float*)d_in[4];
  const float* w3   = (const float*)d_in[5];
  const float* b3   = (const float*)d_in[6];
  const float* sco  = (const float*)d_in[7];
  const float* bio  = (const float*)d_in[8];
  const int*   src  = (const int*)d_in[9];
  const int*   dst  = (const int*)d_in[10];
  // d_in[11] = fn_idx (structurally I..I+F, unused)
  const float* mask = (const float*)d_in[12];
  const int*   last_edge = (const int*)d_in[13];
  const float* has_edge  = (const float*)d_in[14];
  float* out = (float*)d_out;

  char* ws = (char*)d_ws;
  size_t off = 0;
  auto take = [&](size_t bytes) -> void* {
    void* p = ws + off;
    off = (off + bytes + 255) & ~(size_t)255;
    return p;
  };
  float* x0      = (float*)take((size_t)EE * BB * 4);
  float* xe      = (float*)take((size_t)EE * BB * 4);
  float* h       = (float*)take((size_t)FF * BB * CC * 4);  // also h2 (in-place)
  float* w2t     = (float*)take((size_t)FF * CC * CC * 4);
  int*   coff    = (int*)take((size_t)(FF + 1) * 4);
  int*   cnt     = (int*)take((size_t)FF * 4);
  int*   cursor  = (int*)take((size_t)FF * 4);
  int*   edges   = (int*)take((size_t)EE * 4);
  float* partial = (float*)take((size_t)2 * 64 * NB3 * 4);
  float* stats   = (float*)take((size_t)64 * 2 * 4);

  // one-time prep (deterministic: CSR bins sorted after atomic fill)
  k_zero2<<<(FF + 255) / 256, 256, 0, stream>>>(cnt, cursor, FF);
  k_w2t  <<<FF, 256, 0, stream>>>(w2, w2t);
  k_x0   <<<(EE * BB) / 256, 256, 0, stream>>>(x, src, x0, xe);
  k_count<<<(EE + 255) / 256, 256, 0, stream>>>(dst, cnt);
  k_scan <<<1, 256, 0, stream>>>(cnt, coff);
  k_fill <<<(EE + 255) / 256, 256, 0, stream>>>(dst, coff, cursor, edges);
  k_sort <<<(FF + 255) / 256, 256, 0, stream>>>(coff, edges);

  for (int layer = 0; layer < NLAYERS; ++layer) {
    k_h    <<<FF, 256, 0, stream>>>(xe, w1, b1, coff, edges, h);
    k_wmma <<<FF / 8, 256, 0, stream>>>(h, w2t, b2);
    k_edge <<<NB3, 256, 0, stream>>>(h, w3, b3, src, xe, partial);
    k_stats<<<64, 256, 0, stream>>>(partial, stats);
    k_norm <<<(EE * BB) / 256, 256, 0, stream>>>(stats, mask, x0, xe);
  }
  k_out<<<(BB * NN) / 256, 256, 0, stream>>>(xe, sco, bio, last_edge, has_edge, out);
}